// GATConvBlock_21139829031543
// MI455X (gfx1250) — compile-verified
//
#include <hip/hip_runtime.h>
#include <hip/hip_bf16.h>

typedef __attribute__((ext_vector_type(2))) float v2f;
typedef __attribute__((ext_vector_type(8))) float v8f;

#define HEADS    4
#define HEAD_DIM 32
#define CH       128
#define NEG_ATT  0.2f
#define NEG_ACT  0.2f
#define BN_EPS   1e-5f

__device__ __forceinline__ float lrelu(float x, float s) { return x > 0.0f ? x : s * x; }

// Monotone mapping float -> unsigned so unsigned atomicMax == float max.
// +x -> bits | 0x80000000 ; -x -> ~bits.  Sentinel 0 == "no edge arrived".
__device__ __forceinline__ unsigned fmap(float f) {
    unsigned u = __float_as_uint(f);
    return (u & 0x80000000u) ? ~u : (u | 0x80000000u);
}
__device__ __forceinline__ float funmap(unsigned u) {
    return __uint_as_float((u & 0x80000000u) ? (u & 0x7fffffffu) : ~u);
}

// ---------------------------------------------------------------- init
__global__ void k_init(float* __restrict__ out, unsigned* __restrict__ emax,
                       float* __restrict__ denom, float* __restrict__ sums, int N) {
    int idx = blockIdx.x * blockDim.x + threadIdx.x;
    if (idx < N * CH) out[idx] = 0.0f;
    if (idx < N * HEADS) { emax[idx] = 0u; denom[idx] = 0.0f; }
    if (idx < 2 * CH) sums[idx] = 0.0f;
}

// ---------------------------------------------------------------- GEMM: h = x @ W  (fp32 WMMA)
// One wave computes a 16x128 output strip (8 column tiles of 16x16).
__global__ void __launch_bounds__(256) k_gemm(const float* __restrict__ x,
                                              const float* __restrict__ W,
                                              float* __restrict__ h, int N) {
    int ntiles = (N + 15) >> 4;
    int tile = blockIdx.x * 8 + (threadIdx.x >> 5);
    if (tile >= ntiles) return;                 // wave-uniform: EXEC stays all-1s for WMMA

    int lane  = threadIdx.x & 31;
    int lrow  = lane & 15;                      // M (for A) / N column (for B/C)
    int khalf = lane >> 4;                      // 0: K={k,k+1}   1: K={k+2,k+3}
    int row0  = tile << 4;

    int rload = row0 + lrow;  if (rload > N - 1) rload = N - 1;
    const float* xrow = x + (size_t)rload * CH;

    v8f acc[8] = {};
    for (int k = 0; k < CH; k += 4) {
        // A fragment per ISA 32-bit 16x4 layout
        v2f a = *(const v2f*)(xrow + k + (khalf << 1));
        const float* wbase = W + (size_t)(k + (khalf << 1)) * CH + lrow;
        #pragma unroll
        for (int nt = 0; nt < 8; ++nt) {
            v2f b;
            b.x = wbase[nt * 16];               // row k+2*khalf,   col nt*16+lrow
            b.y = wbase[nt * 16 + CH];          // row k+2*khalf+1, col nt*16+lrow
            acc[nt] = __builtin_amdgcn_wmma_f32_16x16x4_f32(
                false, a, false, b, (short)0, acc[nt], false, false);
        }
    }
    // C layout: VGPR r -> M=r (lanes 0-15), M=r+8 (lanes 16-31); N = lrow
    #pragma unroll
    for (int nt = 0; nt < 8; ++nt) {
        #pragma unroll
        for (int r = 0; r < 8; ++r) {
            int row = row0 + r + (khalf << 3);
            if (row < N) h[(size_t)row * CH + nt * 16 + lrow] = acc[nt][r];
        }
    }
}

// ---------------------------------------------------------------- per-node attention coefficients
__global__ void k_att(const float* __restrict__ h, const float* __restrict__ att_src,
                      const float* __restrict__ att_dst,
                      float* __restrict__ a_src, float* __restrict__ a_dst, int N) {
    int idx = blockIdx.x * blockDim.x + threadIdx.x;
    if (idx >= N * HEADS) return;
    int hd = idx & (HEADS - 1);
    const float* hp = h + (size_t)(idx >> 2) * CH + hd * HEAD_DIM;
    const float* as = att_src + hd * HEAD_DIM;
    const float* ad = att_dst + hd * HEAD_DIM;
    float s = 0.0f, d = 0.0f;
    #pragma unroll
    for (int c = 0; c < HEAD_DIM; ++c) { float v = hp[c]; s += v * as[c]; d += v * ad[c]; }
    a_src[idx] = s;
    a_dst[idx] = d;
}

// ---------------------------------------------------------------- edge pass 1: segment max
__global__ void k_edge_max(const long long* __restrict__ ei,
                           const float* __restrict__ a_src, const float* __restrict__ a_dst,
                           unsigned* __restrict__ emax, int E) {
    int idx = blockIdx.x * blockDim.x + threadIdx.x;
    if (idx >= E * HEADS) return;
    int e = idx >> 2, hd = idx & (HEADS - 1);
    long long s = ei[e], d = ei[E + e];
    float sc = lrelu(a_src[s * HEADS + hd] + a_dst[d * HEADS + hd], NEG_ATT);
    atomicMax(&emax[d * HEADS + hd], fmap(sc));
}

// ---------------------------------------------------------------- decode max (in place, isfinite fixup)
__global__ void k_decode(unsigned* __restrict__ emax, int N) {
    int idx = blockIdx.x * blockDim.x + threadIdx.x;
    if (idx >= N * HEADS) return;
    unsigned u = emax[idx];
    ((float*)emax)[idx] = (u == 0u) ? 0.0f : funmap(u);   // no-edge nodes: e_max := 0
}

// ---------------------------------------------------------------- edge pass 2: softmax denominator
__global__ void k_edge_sum(const long long* __restrict__ ei,
                           const float* __restrict__ a_src, const float* __restrict__ a_dst,
                           const float* __restrict__ emax, float* __restrict__ denom, int E) {
    int idx = blockIdx.x * blockDim.x + threadIdx.x;
    if (idx >= E * HEADS) return;
    int e = idx >> 2, hd = idx & (HEADS - 1);
    long long s = ei[e], d = ei[E + e];
    float sc = lrelu(a_src[s * HEADS + hd] + a_dst[d * HEADS + hd], NEG_ATT);
    atomicAdd(&denom[d * HEADS + hd], expf(sc - emax[d * HEADS + hd]));
}

// ---------------------------------------------------------------- edge pass 3: alpha-weighted scatter
__global__ void k_edge_aggr(const long long* __restrict__ ei,
                            const float* __restrict__ a_src, const float* __restrict__ a_dst,
                            const float* __restrict__ emax, const float* __restrict__ denom,
                            const float* __restrict__ h, float* __restrict__ out, int E) {
    long long idx = (long long)blockIdx.x * blockDim.x + threadIdx.x;
    if (idx >= (long long)E * CH) return;
    int e  = (int)(idx >> 7);
    int hc = (int)(idx & (CH - 1));
    int hd = hc >> 5;
    long long s = ei[e], d = ei[E + e];
    float sc    = lrelu(a_src[s * HEADS + hd] + a_dst[d * HEADS + hd], NEG_ATT);
    float ex    = expf(sc - emax[d * HEADS + hd]);
    float alpha = ex / (denom[d * HEADS + hd] + 1e-16f);
    atomicAdd(&out[d * CH + hc], alpha * h[s * CH + hc]);
}

// ---------------------------------------------------------------- BN statistics (column sums)
__global__ void k_bn_stats(const float* __restrict__ out, float* __restrict__ sums,
                           int N, int rowsPerBlock) {
    int c  = threadIdx.x;                       // 128 threads: one channel each
    int r0 = blockIdx.x * rowsPerBlock;
    int r1 = r0 + rowsPerBlock; if (r1 > N) r1 = N;
    float s = 0.0f, ss = 0.0f;
    for (int r = r0; r < r1; ++r) { float v = out[(size_t)r * CH + c]; s += v; ss += v * v; }
    atomicAdd(&sums[c], s);
    atomicAdd(&sums[CH + c], ss);
}

// ---------------------------------------------------------------- BN normalize + LeakyReLU
__global__ void k_bn_act(float* __restrict__ out, const float* __restrict__ sums,
                         const float* __restrict__ gamma, const float* __restrict__ beta, int N) {
    int idx = blockIdx.x * blockDim.x + threadIdx.x;
    if (idx >= N * CH) return;
    int c = idx & (CH - 1);
    float invN = 1.0f / (float)N;
    float mu   = sums[c] * invN;
    float var  = sums[CH + c] * invN - mu * mu;
    float y    = (out[idx] - mu) * rsqrtf(var + BN_EPS) * gamma[c] + beta[c];
    out[idx]   = lrelu(y, NEG_ACT);
}

// ================================================================ launcher
extern "C" void kernel_launch(void* const* d_in, const int* in_sizes, int n_in,
                              void* d_out, int out_size, void* d_ws, size_t ws_size,
                              hipStream_t stream) {
    const float*     x       = (const float*)d_in[0];
    const long long* ei      = (const long long*)d_in[1];   // int64 edge_index [2, E]
    const float*     W       = (const float*)d_in[2];
    const float*     att_src = (const float*)d_in[3];
    const float*     att_dst = (const float*)d_in[4];
    // d_in[5] = bias: cancels exactly under BatchNorm -> skipped
    const float*     gamma   = (const float*)d_in[6];
    const float*     beta    = (const float*)d_in[7];
    float*           out     = (float*)d_out;

    int N = in_sizes[0] / CH;
    int E = in_sizes[1] / 2;

    float*    h     = (float*)d_ws;                              // N*128
    float*    a_src = h + (size_t)N * CH;                        // N*4
    float*    a_dst = a_src + (size_t)N * HEADS;                 // N*4
    unsigned* emax  = (unsigned*)(a_dst + (size_t)N * HEADS);    // N*4 (uint, then float in place)
    float*    denom = (float*)(emax + (size_t)N * HEADS);        // N*4
    float*    sums  = denom + (size_t)N * HEADS;                 // 256

    const int T = 256;
    k_init<<<(N * CH + T - 1) / T, T, 0, stream>>>(out, emax, denom, sums, N);

    int ntiles = (N + 15) / 16;
    k_gemm<<<(ntiles + 7) / 8, 256, 0, stream>>>(x, W, h, N);

    k_att<<<(N * HEADS + T - 1) / T, T, 0, stream>>>(h, att_src, att_dst, a_src, a_dst, N);
    k_edge_max<<<(E * HEADS + T - 1) / T, T, 0, stream>>>(ei, a_src, a_dst, emax, E);
    k_decode<<<(N * HEADS + T - 1) / T, T, 0, stream>>>(emax, N);
    k_edge_sum<<<(E * HEADS + T - 1) / T, T, 0, stream>>>(ei, a_src, a_dst, (const float*)emax, denom, E);

    long long tot = (long long)E * CH;
    k_edge_aggr<<<(unsigned)((tot + T - 1) / T), T, 0, stream>>>(
        ei, a_src, a_dst, (const float*)emax, denom, h, out, E);

    int nb = 512, rpb = (N + nb - 1) / nb;
    k_bn_stats<<<nb, CH, 0, stream>>>(out, sums, N, rpb);
    k_bn_act<<<(N * CH + T - 1) / T, T, 0, stream>>>(out, sums, gamma, beta, N);
}